// GCNModel_87402584473801
// MI455X (gfx1250) — compile-verified
//
#include <hip/hip_runtime.h>
#include <hip/hip_bf16.h>

// ---------------------------------------------------------------------------
// GCN forward for MI455X (gfx1250, wave32).
// GEMMs: V_WMMA_F32_16X16X4_F32. Weights staged in LDS pre-paired by K so each
// B fragment is ONE aligned ds_load_b64 (no repack moves), bank-conflict-free
// (RS=160: half-wave offset 32 banks). K fully unrolled -> immediate offsets.
// Edge aggregation: global_atomic_add_f32 (this is the bandwidth floor).
// ---------------------------------------------------------------------------

typedef __attribute__((ext_vector_type(2))) float v2f;
typedef __attribute__((ext_vector_type(8))) float v8f;

#define DIN  128
#define DH   64
#define DOUT 40
#define LNEPS 1e-5f
#define RS   160   // LDS dwords per K-pair row; RS % 64 == 32 -> disjoint banks

__device__ __forceinline__ int iclamp(int v, int hi) { return v < hi ? v : hi; }

__device__ __forceinline__ v8f wmma4(v2f a, v2f b, v8f c) {
    return __builtin_amdgcn_wmma_f32_16x16x4_f32(
        /*neg_a=*/false, a, /*neg_b=*/false, b,
        /*c_mod=*/(short)0, c, /*reuse_a=*/false, /*reuse_b=*/false);
}

// ---------------------------------------------------------------------------
// Elementwise / degree kernels
// ---------------------------------------------------------------------------
__global__ void k_fill(float* p, float v, int n) {
    int i = blockIdx.x * blockDim.x + threadIdx.x;
    if (i < n) p[i] = v;
}

__global__ void k_deg_count(const int* __restrict__ dst, float* __restrict__ deg, int E) {
    int e = blockIdx.x * blockDim.x + threadIdx.x;
    if (e < E) atomicAdd(&deg[dst[e]], 1.0f);
}

__global__ void k_rsqrt_inplace(float* p, int n) {
    int i = blockIdx.x * blockDim.x + threadIdx.x;
    if (i < n) p[i] = rsqrtf(p[i]);
}

__global__ void k_norm_e(const int* __restrict__ src, const int* __restrict__ dst,
                         const float* __restrict__ dinv, float* __restrict__ norm_e, int E) {
    int e = blockIdx.x * blockDim.x + threadIdx.x;
    if (e < E) norm_e[e] = dinv[src[e]] * dinv[dst[e]];
}

// agg[i][c..c+3] = g[i][c..c+3] * dinv[i]^2 + bias[c..c+3]  (self-loop + bias)
__global__ void k_agg_init(const float4* __restrict__ g, const float* __restrict__ dinv,
                           const float4* __restrict__ bias, float4* __restrict__ agg, int total4) {
    int i = blockIdx.x * blockDim.x + threadIdx.x;
    if (i < total4) {
        int node = i >> 4;           // 16 float4 per 64-channel row
        int c4   = i & 15;
        float w  = dinv[node];
        w *= w;
        float4 gv = g[i];
        float4 bv = bias[c4];
        float4 o;
        o.x = gv.x * w + bv.x;
        o.y = gv.y * w + bv.y;
        o.z = gv.z * w + bv.z;
        o.w = gv.w * w + bv.w;
        agg[i] = o;
    }
}

// agg[dst][c] += norm_e[e] * g[src][c]  — one thread per (edge, channel pair)
__global__ void k_scatter(const int* __restrict__ src, const int* __restrict__ dst,
                          const float* __restrict__ norm_e, const float* __restrict__ g,
                          float* __restrict__ agg, long long total) {
    long long i = (long long)blockIdx.x * blockDim.x + threadIdx.x;
    if (i < total) {
        int e = (int)(i >> 5);
        int c = (int)(i & 31) * 2;
        int s = src[e];
        int d = dst[e];
        float w = norm_e[e];
        v2f gv = *(const v2f*)(g + (long long)s * 64 + c);
        float* ap = agg + (long long)d * 64 + c;
        atomicAdd(ap + 0, w * gv.x);
        atomicAdd(ap + 1, w * gv.y);
    }
}

__global__ void k_relu(float* p, int n) {
    int i = blockIdx.x * blockDim.x + threadIdx.x;
    if (i < n) p[i] = fmaxf(p[i], 0.0f);
}

// ReLU then LayerNorm over the 64-channel row; one wave32 per row, 2 elems/lane.
__global__ void k_relu_ln(float* __restrict__ h, const float* __restrict__ gamma,
                          const float* __restrict__ beta, int n) {
    int wid  = (blockIdx.x * blockDim.x + threadIdx.x) >> 5;   // wave id == row
    int lane = threadIdx.x & 31;
    if (wid >= n) return;                                      // wave-uniform exit
    float* row = h + (long long)wid * 64;
    float v0 = fmaxf(row[lane],      0.0f);
    float v1 = fmaxf(row[lane + 32], 0.0f);
    float s = v0 + v1;
    #pragma unroll
    for (int m = 16; m >= 1; m >>= 1) s += __shfl_xor(s, m, 32);
    float mu = s * (1.0f / 64.0f);
    float d0 = v0 - mu, d1 = v1 - mu;
    float q = d0 * d0 + d1 * d1;
    #pragma unroll
    for (int m = 16; m >= 1; m >>= 1) q += __shfl_xor(q, m, 32);
    float inv = rsqrtf(q * (1.0f / 64.0f) + LNEPS);
    row[lane]      = d0 * inv * gamma[lane]      + beta[lane];
    row[lane + 32] = d1 * inv * gamma[lane + 32] + beta[lane + 32];
}

// ---------------------------------------------------------------------------
// fp32 WMMA GEMM:  C[M x Nout] = A[M x K] * W[K x Nout] (+ bias), Nout <= 64.
// LDS layout: K-paired —  wlds[(k>>1)*RS + 2*c + (k&1)] = W[k][c] (zero-padded
// cols >= Nout).  A lane's B fragment {W[k+2h][c], W[k+2h+1][c]} is then ONE
// aligned ds_load_b64 at  base(half*RS + 2*lm) + k*(RS/2) + 32*t  elements.
// Per-lane mapping (ISA 7.12.2), half=lane>>4, lm=lane&15:
//   A frag: a[r] = A[row0+lm][k + 2*half + r]   (global_load_b64, imm offsets)
//   B frag: b[r] = W[k + 2*half + r][t*16 + lm] (ds_load_b64, imm offsets)
//   D frag: c[r] = C[row0 + r + 8*half][t*16 + lm]
// ---------------------------------------------------------------------------
template <int K, bool BIAS>
__global__ void gemm_f32_wmma(const float* __restrict__ A, const float* __restrict__ W,
                              const float* __restrict__ bias, float* __restrict__ C,
                              int M, int Nout) {
    __shared__ float wlds[(K / 2) * RS];

    // Cooperative fill: logical K x 64 tile, zero-padded, K-paired layout.
    for (int idx = threadIdx.x; idx < K * 64; idx += 256) {
        int r = idx >> 6;
        int c = idx & 63;
        float v = (c < Nout) ? W[r * Nout + c] : 0.f;
        wlds[(r >> 1) * RS + 2 * c + (r & 1)] = v;
    }
    __syncthreads();

    int wave = (blockIdx.x * blockDim.x + threadIdx.x) >> 5;
    int lane = threadIdx.x & 31;
    int row0 = wave * 16;
    if (row0 >= M) return;                        // wave-uniform (after barrier)

    int half = lane >> 4;
    int lm   = lane & 15;
    int arow = iclamp(row0 + lm, M - 1);          // clamp instead of predicate
    const float* Ap = A + (long long)arow * K + 2 * half;
    const float* wl = wlds + half * RS + 2 * lm;  // per-lane LDS base (8B aligned)

    const v8f zero8 = {0.f, 0.f, 0.f, 0.f, 0.f, 0.f, 0.f, 0.f};
    v8f acc0 = zero8, acc1 = zero8, acc2 = zero8, acc3 = zero8;

    #pragma unroll
    for (int k = 0; k < K; k += 4) {
        v2f a = *(const v2f*)(Ap + k);
        const float* wk = wl + k * (RS / 2);      // j = k/2 + half  ->  k*RS/2
        v2f b0 = *(const v2f*)(wk + 0);
        v2f b1 = *(const v2f*)(wk + 32);
        v2f b2 = *(const v2f*)(wk + 64);
        v2f b3 = *(const v2f*)(wk + 96);
        acc0 = wmma4(a, b0, acc0);
        acc1 = wmma4(a, b1, acc1);
        acc2 = wmma4(a, b2, acc2);
        acc3 = wmma4(a, b3, acc3);
    }

    auto store_tile = [&](const v8f& acc, int t) {
        int col = t * 16 + lm;
        if (col < Nout) {
            float bv = BIAS ? bias[col] : 0.f;
            #pragma unroll
            for (int r = 0; r < 8; ++r) {
                int row = row0 + r + 8 * half;
                if (row < M) C[(long long)row * Nout + col] = acc[r] + bv;
            }
        }
    };
    store_tile(acc0, 0);
    store_tile(acc1, 1);
    store_tile(acc2, 2);
    store_tile(acc3, 3);
}

// ---------------------------------------------------------------------------
// Launcher
// ---------------------------------------------------------------------------
extern "C" void kernel_launch(void* const* d_in, const int* in_sizes, int n_in,
                              void* d_out, int out_size, void* d_ws, size_t ws_size,
                              hipStream_t stream) {
    // Inputs per setup_inputs() order:
    const float* x      = (const float*)d_in[0];    // [N, 128]
    const int*   eidx   = (const int*)  d_in[1];    // [2, E]
    const float* W0     = (const float*)d_in[2];
    const float* b0     = (const float*)d_in[3];
    const float* W1     = (const float*)d_in[4];
    const float* b1     = (const float*)d_in[5];
    const float* W2     = (const float*)d_in[6];
    const float* b2     = (const float*)d_in[7];
    const float* ln0_g  = (const float*)d_in[8];
    const float* ln0_b  = (const float*)d_in[9];
    const float* ln1_g  = (const float*)d_in[10];
    const float* ln1_b  = (const float*)d_in[11];
    const float* mpW0   = (const float*)d_in[12];
    const float* mpb0   = (const float*)d_in[13];
    const float* mpW1   = (const float*)d_in[14];
    const float* mpb1   = (const float*)d_in[15];

    const int N = in_sizes[0] / DIN;
    const int E = in_sizes[1] / 2;
    const int* src = eidx;
    const int* dst = eidx + E;

    // Workspace layout (floats):
    //   [0, N)                      deg -> dinv (in place)
    //   [N, N+E)                    norm_e
    //   [N+E, N+E+N*64)             gbuf  (GEMM outputs)
    //   [N+E+N*64, N+E+2*N*64)      abuf  (aggregation / activations)
    float* ws     = (float*)d_ws;
    float* dinv   = ws;
    float* norm_e = ws + N;
    float* gbuf   = norm_e + E;
    float* abuf   = gbuf + (long long)N * 64;
    float* out    = (float*)d_out;

    const int TB = 256;
    const int nodeBlocks = (N + TB - 1) / TB;
    const int edgeBlocks = (E + TB - 1) / TB;
    const long long nc   = (long long)N * 64;
    const int ncBlocks   = (int)((nc + TB - 1) / TB);
    const int nc4Blocks  = (int)((nc / 4 + TB - 1) / TB);
    const long long ec2  = (long long)E * 32;          // (edge, channel-pair)
    const int ec2Blocks  = (int)((ec2 + TB - 1) / TB);
    const int lnBlocks   = (int)(((long long)N * 32 + TB - 1) / TB);
    // GEMM: 8 waves/block, 16 rows per wave -> 128 rows per block
    const int gemmBlocks = (N + 127) / 128;

    // ---- degree / edge norms (computed once, reused by all 3 conv layers) ----
    k_fill<<<nodeBlocks, TB, 0, stream>>>(dinv, 1.0f, N);          // deg = 1 (self loop)
    k_deg_count<<<edgeBlocks, TB, 0, stream>>>(dst, dinv, E);      // deg += in-degree
    k_rsqrt_inplace<<<nodeBlocks, TB, 0, stream>>>(dinv, N);       // deg -> d^{-1/2}
    k_norm_e<<<edgeBlocks, TB, 0, stream>>>(src, dst, dinv, norm_e, E);

    // ---- conv layer 0: x(128) -> 64 ----
    gemm_f32_wmma<DIN, false><<<gemmBlocks, TB, 0, stream>>>(x, W0, nullptr, gbuf, N, DH);
    k_agg_init<<<nc4Blocks, TB, 0, stream>>>((const float4*)gbuf, dinv, (const float4*)b0,
                                             (float4*)abuf, (int)(nc / 4));
    k_scatter<<<ec2Blocks, TB, 0, stream>>>(src, dst, norm_e, gbuf, abuf, ec2);
    k_relu_ln<<<lnBlocks, TB, 0, stream>>>(abuf, ln0_g, ln0_b, N);

    // ---- conv layer 1: 64 -> 64 ----
    gemm_f32_wmma<DH, false><<<gemmBlocks, TB, 0, stream>>>(abuf, W1, nullptr, gbuf, N, DH);
    k_agg_init<<<nc4Blocks, TB, 0, stream>>>((const float4*)gbuf, dinv, (const float4*)b1,
                                             (float4*)abuf, (int)(nc / 4));
    k_scatter<<<ec2Blocks, TB, 0, stream>>>(src, dst, norm_e, gbuf, abuf, ec2);
    k_relu_ln<<<lnBlocks, TB, 0, stream>>>(abuf, ln1_g, ln1_b, N);

    // ---- conv layer 2: 64 -> 64, relu only ----
    gemm_f32_wmma<DH, false><<<gemmBlocks, TB, 0, stream>>>(abuf, W2, nullptr, gbuf, N, DH);
    k_agg_init<<<nc4Blocks, TB, 0, stream>>>((const float4*)gbuf, dinv, (const float4*)b2,
                                             (float4*)abuf, (int)(nc / 4));
    k_scatter<<<ec2Blocks, TB, 0, stream>>>(src, dst, norm_e, gbuf, abuf, ec2);
    k_relu<<<ncBlocks, TB, 0, stream>>>(abuf, (int)nc);

    // ---- post_mp: Linear(64->64)+bias, then Linear(64->40)+bias ----
    gemm_f32_wmma<DH, true><<<gemmBlocks, TB, 0, stream>>>(abuf, mpW0, mpb0, gbuf, N, DH);
    gemm_f32_wmma<DH, true><<<gemmBlocks, TB, 0, stream>>>(gbuf, mpW1, mpb1, out, N, DOUT);
}